// LightGCN_29291676959275
// MI455X (gfx1250) — compile-verified
//
#include <hip/hip_runtime.h>

// LightGCN 2-layer propagate: h0 = segsum(x[src0]*ew0 -> dst0, 50000)
//                             out = segsum(h0[src1]*ew1 -> dst1, 25000)
// D = 128 floats per row. Memory/atomic bound => no WMMA; use CDNA5
// async-to-LDS staging + prefetch + native f32 global atomics.

#define WAVES_PER_BLOCK 8
#define TILE 64

__global__ __launch_bounds__(256) void lgcn_zero(float* __restrict__ p, long long n) {
  long long i = ((long long)blockIdx.x * blockDim.x + threadIdx.x) * 4;
  long long stride = (long long)gridDim.x * blockDim.x * 4;
  float4 z = make_float4(0.f, 0.f, 0.f, 0.f);
  for (; i < n; i += stride) *(float4*)(p + i) = z;
}

__global__ __launch_bounds__(256) void lgcn_propagate(
    const float* __restrict__ hin, const int* __restrict__ src,
    const int* __restrict__ dst, const float* __restrict__ ew,
    float* __restrict__ out, int E) {
  // Per-wave double-buffered staging area for edge triples.
  __shared__ int   sm_src[WAVES_PER_BLOCK][2][TILE];
  __shared__ int   sm_dst[WAVES_PER_BLOCK][2][TILE];
  __shared__ float sm_w  [WAVES_PER_BLOCK][2][TILE];

  const int lane = threadIdx.x & 31;
  const int wv   = threadIdx.x >> 5;

  const long long stride = (long long)gridDim.x * WAVES_PER_BLOCK * TILE;
  long long base = ((long long)blockIdx.x * WAVES_PER_BLOCK + wv) * TILE;
  if (base >= E) return;

  // Stage 64 edges' (src,dst,ew) into LDS buffer b via async load-to-LDS.
  // Lane l handles edges pos+l and pos+32+l (addresses clamped to stay in range;
  // clamped slots are never consumed because the process loop bounds by jmax).
  auto stage = [&](int b, long long pos) {
    long long eA = pos + lane;      if (eA >= E) eA = E - 1;
    long long eB = pos + 32 + lane; if (eB >= E) eB = E - 1;
    unsigned lsA = (unsigned)(size_t)&sm_src[wv][b][lane];
    unsigned lsB = (unsigned)(size_t)&sm_src[wv][b][lane + 32];
    unsigned ldA = (unsigned)(size_t)&sm_dst[wv][b][lane];
    unsigned ldB = (unsigned)(size_t)&sm_dst[wv][b][lane + 32];
    unsigned lwA = (unsigned)(size_t)&sm_w[wv][b][lane];
    unsigned lwB = (unsigned)(size_t)&sm_w[wv][b][lane + 32];
    asm volatile("global_load_async_to_lds_b32 %0, %1, off"
                 :: "v"(lsA), "v"((unsigned long long)(size_t)(src + eA)) : "memory");
    asm volatile("global_load_async_to_lds_b32 %0, %1, off"
                 :: "v"(lsB), "v"((unsigned long long)(size_t)(src + eB)) : "memory");
    asm volatile("global_load_async_to_lds_b32 %0, %1, off"
                 :: "v"(ldA), "v"((unsigned long long)(size_t)(dst + eA)) : "memory");
    asm volatile("global_load_async_to_lds_b32 %0, %1, off"
                 :: "v"(ldB), "v"((unsigned long long)(size_t)(dst + eB)) : "memory");
    asm volatile("global_load_async_to_lds_b32 %0, %1, off"
                 :: "v"(lwA), "v"((unsigned long long)(size_t)(ew + eA)) : "memory");
    asm volatile("global_load_async_to_lds_b32 %0, %1, off"
                 :: "v"(lwB), "v"((unsigned long long)(size_t)(ew + eB)) : "memory");
  };

  int cur = 0;
  stage(cur, base);

  for (; base < E; base += stride) {
    long long nxt = base + stride;
    if (nxt < E) {
      stage(cur ^ 1, nxt);                              // overlap next tile's metadata fetch
      asm volatile("s_wait_asynccnt 6" ::: "memory");   // current tile's 6 transfers done
    } else {
      asm volatile("s_wait_asynccnt 0" ::: "memory");
    }

    int jmax = (E - base < (long long)TILE) ? (int)(E - base) : TILE;

#pragma unroll 4
    for (int j = 0; j < jmax; ++j) {
      int   s = sm_src[wv][cur][j];   // LDS broadcast reads
      int   d = sm_dst[wv][cur][j];
      float w = sm_w[wv][cur][j];

      // Look-ahead prefetch: pull the row needed 8 edges from now into cache.
      if (j + 8 < jmax) {
        int sp = sm_src[wv][cur][j + 8];
        __builtin_prefetch(hin + (size_t)sp * 128 + lane * 4, 0, 0);
      }

      // Gather: one wave covers the whole 512B row, float4 per lane.
      const float4* __restrict__ row = (const float4*)(hin + (size_t)s * 128);
      float4 v = row[lane];

      // Scatter-accumulate: native no-return f32 atomics (4B each, 16B/lane).
      unsigned long long oa =
          (unsigned long long)(size_t)(out + (size_t)d * 128 + lane * 4);
      asm volatile("global_atomic_add_f32 %0, %1, off"           :: "v"(oa), "v"(v.x * w));
      asm volatile("global_atomic_add_f32 %0, %1, off offset:4"  :: "v"(oa), "v"(v.y * w));
      asm volatile("global_atomic_add_f32 %0, %1, off offset:8"  :: "v"(oa), "v"(v.z * w));
      asm volatile("global_atomic_add_f32 %0, %1, off offset:12" :: "v"(oa), "v"(v.w * w));
    }
    cur ^= 1;
  }
}

static inline int blocks_for(int E) {
  // ~4 grid-stride sweeps per wave so the async double-buffer actually pipelines.
  long long b = ((long long)E + (long long)(WAVES_PER_BLOCK * TILE * 4) - 1) /
                (long long)(WAVES_PER_BLOCK * TILE * 4);
  if (b < 1) b = 1;
  if (b > 4096) b = 4096;
  return (int)b;
}

extern "C" void kernel_launch(void* const* d_in, const int* in_sizes, int n_in,
                              void* d_out, int out_size, void* d_ws, size_t ws_size,
                              hipStream_t stream) {
  (void)n_in; (void)ws_size;
  const float* x    = (const float*)d_in[0];
  const int*   src0 = (const int*)d_in[1];
  const int*   dst0 = (const int*)d_in[2];
  const float* ew0  = (const float*)d_in[3];
  const int*   src1 = (const int*)d_in[4];
  const int*   dst1 = (const int*)d_in[5];
  const float* ew1  = (const float*)d_in[6];

  const int E0 = in_sizes[1];
  const int E1 = in_sizes[4];
  const long long NDST0 = 50000;      // reference constant (n_dst0 arrives as device scalar)
  const int D = 128;

  float* h0  = (float*)d_ws;          // 50000*128*4 = 25.6 MB scratch for layer-0 output
  float* out = (float*)d_out;

  // Zero accumulators every call (buffers are poisoned / carry state otherwise).
  lgcn_zero<<<dim3(2048), dim3(256), 0, stream>>>(h0, NDST0 * D);
  lgcn_zero<<<dim3(1024), dim3(256), 0, stream>>>(out, (long long)out_size);

  lgcn_propagate<<<dim3(blocks_for(E0)), dim3(256), 0, stream>>>(x,  src0, dst0, ew0, h0,  E0);
  lgcn_propagate<<<dim3(blocks_for(E1)), dim3(256), 0, stream>>>(h0, src1, dst1, ew1, out, E1);
}